// BA3MotifNet_67602785239192
// MI455X (gfx1250) — compile-verified
//
#include <hip/hip_runtime.h>

// ---------------------------------------------------------------------------
// BA3MotifNet forward on gfx1250 (MI455X).
// fp32 throughout; GEMMs use V_WMMA_F32_16X16X4_F32 (native fp32 WMMA).
// Memory-bound edge gather/scatter uses coalesced float4 + global f32 atomics.
// ---------------------------------------------------------------------------

typedef __attribute__((ext_vector_type(2))) float v2f;
typedef __attribute__((ext_vector_type(8))) float v8f;

#define HID 64

// ---------------------------------------------------------------------------
// h = x @ emb_w + emb_b      x:[N,4]  emb_w:[4,64]
// one thread per (node, j); coalesced on j.
// ---------------------------------------------------------------------------
__global__ __launch_bounds__(256)
void embed_kernel(const float* __restrict__ x,
                  const float* __restrict__ emb_w,
                  const float* __restrict__ emb_b,
                  float* __restrict__ h, int nNodes)
{
    int gid = blockIdx.x * blockDim.x + threadIdx.x;
    int n = gid >> 6;          // /64
    int j = gid & 63;
    if (n >= nNodes) return;
    const float* xr = x + n * 4;
    float acc = emb_b[j];
    acc += xr[0] * emb_w[0 * HID + j];
    acc += xr[1] * emb_w[1 * HID + j];
    acc += xr[2] * emb_w[2 * HID + j];
    acc += xr[3] * emb_w[3 * HID + j];
    h[n * HID + j] = acc;
}

// ---------------------------------------------------------------------------
// Fused triple GEMM via fp32 WMMA:
//   outA = h@W1 + b1 ; outB = h@W2 ; outC = h@W3 + b3
// Each wave: one 16-node tile, loops over 4 column tiles of 16.
// K = 64 -> 16 steps of V_WMMA_F32_16X16X4_F32 per (tile, mat).
//
// A fragment (16x4 f32, 2 VGPRs/lane):
//   lane<16 : M=lane,    holds K = 4kk+0, 4kk+1
//   lane>=16: M=lane-16, holds K = 4kk+2, 4kk+3
// B fragment (4x16 f32): same K split across lane halves, N = lane&15.
// C/D (16x16 f32, 8 VGPRs): elem v -> row v + 8*(lane>>4), col = lane&15.
// ---------------------------------------------------------------------------
__global__ __launch_bounds__(256)
void gemm3_kernel(const float* __restrict__ h,
                  const float* __restrict__ W1, const float* __restrict__ B1,
                  const float* __restrict__ W2,
                  const float* __restrict__ W3, const float* __restrict__ B3,
                  float* __restrict__ outA,
                  float* __restrict__ outB,
                  float* __restrict__ outC,
                  int nNodes)
{
    const int lane  = threadIdx.x & 31;
    const int wave  = threadIdx.x >> 5;
    const int tile  = blockIdx.x * 8 + wave;          // 16-node tile (wave-uniform)
    const int row0  = tile * 16;
    if (row0 >= nNodes) return;                       // uniform exit: EXEC stays all-1s

    const int m     = lane & 15;
    const int khalf = lane >> 4;                      // 0 or 1

    // Preload all 16 A fragments for this node tile (reused for all col tiles)
    int row = row0 + m;
    if (row >= nNodes) row = nNodes - 1;              // safety clamp (N%16==0 normally)
    v2f afrag[16];
#pragma unroll
    for (int kk = 0; kk < 16; ++kk) {
        const float* p = h + row * HID + kk * 4 + khalf * 2;
        afrag[kk] = v2f{p[0], p[1]};
    }

    for (int ct = 0; ct < 4; ++ct) {
        const int j = ct * 16 + m;                    // output column
        v8f acc1 = {}, acc2 = {}, acc3 = {};
        const float b1 = B1[j];
        const float b3 = B3[j];
#pragma unroll
        for (int v = 0; v < 8; ++v) { acc1[v] = b1; acc3[v] = b3; }

#pragma unroll
        for (int kk = 0; kk < 16; ++kk) {
            const int kb = kk * 4 + khalf * 2;
            v2f w1 = v2f{W1[kb * HID + j], W1[(kb + 1) * HID + j]};
            v2f w2 = v2f{W2[kb * HID + j], W2[(kb + 1) * HID + j]};
            v2f w3 = v2f{W3[kb * HID + j], W3[(kb + 1) * HID + j]};
            acc1 = __builtin_amdgcn_wmma_f32_16x16x4_f32(
                       false, afrag[kk], false, w1, (short)0, acc1, false, false);
            acc2 = __builtin_amdgcn_wmma_f32_16x16x4_f32(
                       false, afrag[kk], false, w2, (short)0, acc2, false, false);
            acc3 = __builtin_amdgcn_wmma_f32_16x16x4_f32(
                       false, afrag[kk], false, w3, (short)0, acc3, false, false);
        }

#pragma unroll
        for (int v = 0; v < 8; ++v) {
            int r = row0 + v + khalf * 8;
            if (r < nNodes) {
                outA[r * HID + j] = acc1[v];
                outB[r * HID + j] = acc2[v];
                outC[r * HID + j] = acc3[v];   // agg buffer seeded with h@W3+b3
            }
        }
    }
}

// ---------------------------------------------------------------------------
// Edge phase: msg = (a[src] - b[dst]) * ew;  agg[dst] += msg (atomic f32).
// 16 lanes per edge, float4 per lane -> coalesced 256B gathers per edge row.
// ---------------------------------------------------------------------------
__global__ __launch_bounds__(256)
void edge_kernel(const int* __restrict__ srcIdx,
                 const int* __restrict__ dstIdx,
                 const float* __restrict__ ew,
                 const float* __restrict__ A,
                 const float* __restrict__ B,
                 float* __restrict__ agg, int nEdges)
{
    int gid = blockIdx.x * blockDim.x + threadIdx.x;
    int e = gid >> 4;                                 // edge
    int c = gid & 15;                                 // float4 chunk
    if (e >= nEdges) return;
    int s = srcIdx[e];
    int d = dstIdx[e];
    float w = ew[e];
    const float4 va = *(const float4*)(A + s * HID + c * 4);
    const float4 vb = *(const float4*)(B + d * HID + c * 4);
    float* p = agg + d * HID + c * 4;
    atomicAdd(p + 0, (va.x - vb.x) * w);
    atomicAdd(p + 1, (va.y - vb.y) * w);
    atomicAdd(p + 2, (va.z - vb.z) * w);
    atomicAdd(p + 3, (va.w - vb.w) * w);
}

// ---------------------------------------------------------------------------
// h = relu(agg)
// ---------------------------------------------------------------------------
__global__ __launch_bounds__(256)
void relu_kernel(const float* __restrict__ agg, float* __restrict__ h, int n)
{
    int gid = blockIdx.x * blockDim.x + threadIdx.x;
    if (gid >= n) return;
    h[gid] = fmaxf(agg[gid], 0.0f);
}

// ---------------------------------------------------------------------------
// zero sums/cnt
// ---------------------------------------------------------------------------
__global__ __launch_bounds__(256)
void zero_kernel(float* __restrict__ p, int n)
{
    int gid = blockIdx.x * blockDim.x + threadIdx.x;
    if (gid < n) p[gid] = 0.0f;
}

// ---------------------------------------------------------------------------
// Graph pooling: sums[batch[n]] += h[n]; cnt[g] += 1.
// ---------------------------------------------------------------------------
__global__ __launch_bounds__(256)
void pool_kernel(const float* __restrict__ h,
                 const int* __restrict__ batch,
                 float* __restrict__ sums,
                 float* __restrict__ cnt, int nNodes)
{
    int gid = blockIdx.x * blockDim.x + threadIdx.x;
    int n = gid >> 4;
    int c = gid & 15;
    if (n >= nNodes) return;
    int g = batch[n];
    const float4 v = *(const float4*)(h + n * HID + c * 4);
    float* p = sums + g * HID + c * 4;
    atomicAdd(p + 0, v.x);
    atomicAdd(p + 1, v.y);
    atomicAdd(p + 2, v.z);
    atomicAdd(p + 3, v.w);
    if (c == 0) atomicAdd(cnt + g, 1.0f);
}

// ---------------------------------------------------------------------------
// MLP head: pred = relu(gx @ lin1_w + b1) @ lin2_w + b2   (one block / graph)
// ---------------------------------------------------------------------------
__global__ __launch_bounds__(64)
void head_kernel(const float* __restrict__ sums,
                 const float* __restrict__ cnt,
                 const float* __restrict__ l1w, const float* __restrict__ l1b,
                 const float* __restrict__ l2w, const float* __restrict__ l2b,
                 float* __restrict__ out)
{
    __shared__ float gx[HID];
    __shared__ float y1[HID];
    const int g = blockIdx.x;
    const int t = threadIdx.x;
    const float cinv = 1.0f / fmaxf(cnt[g], 1.0f);
    gx[t] = sums[g * HID + t] * cinv;
    __syncthreads();
    float acc = l1b[t];
#pragma unroll 8
    for (int k = 0; k < HID; ++k) acc += gx[k] * l1w[k * HID + t];
    y1[t] = fmaxf(acc, 0.0f);
    __syncthreads();
    if (t < 3) {
        float a = l2b[t];
#pragma unroll 8
        for (int k = 0; k < HID; ++k) a += y1[k] * l2w[k * 3 + t];
        out[g * 3 + t] = a;
    }
}

// ---------------------------------------------------------------------------
// Host-side orchestration
// ---------------------------------------------------------------------------
extern "C" void kernel_launch(void* const* d_in, const int* in_sizes, int n_in,
                              void* d_out, int out_size, void* d_ws, size_t ws_size,
                              hipStream_t stream) {
    const float* x     = (const float*)d_in[0];
    const int*   ei    = (const int*)  d_in[1];   // [2, E]
    const float* ea    = (const float*)d_in[2];
    const int*   batch = (const int*)  d_in[3];
    const float* emb_w = (const float*)d_in[4];
    const float* emb_b = (const float*)d_in[5];
    const float* cw1   = (const float*)d_in[6];   // [L,64,64]
    const float* cb1   = (const float*)d_in[7];   // [L,64]
    const float* cw2   = (const float*)d_in[8];
    const float* cw3   = (const float*)d_in[9];
    const float* cb3   = (const float*)d_in[10];
    const float* l1w   = (const float*)d_in[11];
    const float* l1b   = (const float*)d_in[12];
    const float* l2w   = (const float*)d_in[13];
    const float* l2b   = (const float*)d_in[14];
    float* out = (float*)d_out;

    const int N = in_sizes[0] / 4;                 // nodes
    const int E = in_sizes[2];                     // edges
    const int L = in_sizes[6] / (HID * HID);       // layers
    const int G = out_size / 3;                    // graphs

    // workspace layout (floats)
    float* ws   = (float*)d_ws;
    float* h    = ws;                     // N*64
    float* a    = h    + (size_t)N * HID; // N*64
    float* b    = a    + (size_t)N * HID; // N*64
    float* agg  = b    + (size_t)N * HID; // N*64
    float* sums = agg  + (size_t)N * HID; // G*64
    float* cnt  = sums + (size_t)G * HID; // G

    const int* srcIdx = ei;
    const int* dstIdx = ei + E;

    // 1. embedding
    {
        int total = N * HID;
        embed_kernel<<<(total + 255) / 256, 256, 0, stream>>>(x, emb_w, emb_b, h, N);
    }

    // 2. conv layers
    const int tiles  = (N + 15) / 16;
    const int gblocks = (tiles + 7) / 8;
    for (int l = 0; l < L; ++l) {
        const float* W1 = cw1 + (size_t)l * HID * HID;
        const float* W2 = cw2 + (size_t)l * HID * HID;
        const float* W3 = cw3 + (size_t)l * HID * HID;
        const float* B1 = cb1 + (size_t)l * HID;
        const float* B3 = cb3 + (size_t)l * HID;

        gemm3_kernel<<<gblocks, 256, 0, stream>>>(h, W1, B1, W2, W3, B3, a, b, agg, N);

        int etotal = E * 16;
        edge_kernel<<<(etotal + 255) / 256, 256, 0, stream>>>(srcIdx, dstIdx, ea, a, b, agg, E);

        int ntotal = N * HID;
        relu_kernel<<<(ntotal + 255) / 256, 256, 0, stream>>>(agg, h, ntotal);
    }

    // 3. pooling
    {
        int ztotal = G * HID + G;
        zero_kernel<<<(ztotal + 255) / 256, 256, 0, stream>>>(sums, ztotal);  // sums then cnt (contiguous)
        int ptotal = N * 16;
        pool_kernel<<<(ptotal + 255) / 256, 256, 0, stream>>>(h, batch, sums, cnt, N);
    }

    // 4. MLP head
    head_kernel<<<G, 64, 0, stream>>>(sums, cnt, l1w, l1b, l2w, l2b, out);
}